// ECELoss_30734785970356
// MI455X (gfx1250) — compile-verified
//
#include <hip/hip_runtime.h>

#define NBINS    15
#define TPB      192                // 6 wave32s
#define NWAVES   (TPB / 32)
#define HSTRIDE  49                 // per-wave hist stride in floats (odd -> bank spread)
#define CLS      32                 // classes (fixed by reference)
#define ROWPAD   144                // LDS bytes per row: 36 DWORDs -> conflict-free b128 reads
#define BUFBYTES (TPB * ROWPAD)     // 27648 B per tile buffer

// ---- CDNA5 async global->LDS copy (ASYNCcnt path) -------------------------
// ISA: the immediate offset is added to both the LDS and the global address
template <int OFF>
__device__ __forceinline__ void async_load_b128_off(unsigned lds_addr, const void* gaddr) {
    asm volatile("global_load_async_to_lds_b128 %0, %1, off offset:%2"
                 :: "v"(lds_addr), "v"(gaddr), "n"(OFF)
                 : "memory");
}

__device__ __forceinline__ void issue_row_copy(unsigned lds_addr, const void* gaddr) {
    async_load_b128_off<  0>(lds_addr, gaddr);
    async_load_b128_off< 16>(lds_addr, gaddr);
    async_load_b128_off< 32>(lds_addr, gaddr);
    async_load_b128_off< 48>(lds_addr, gaddr);
    async_load_b128_off< 64>(lds_addr, gaddr);
    async_load_b128_off< 80>(lds_addr, gaddr);
    async_load_b128_off< 96>(lds_addr, gaddr);
    async_load_b128_off<112>(lds_addr, gaddr);
}

template <int N>
__device__ __forceinline__ void wait_asynccnt() {
    asm volatile("s_wait_asynccnt %0" :: "n"(N) : "memory");
}

__device__ __forceinline__ float fast_exp2(float x) {
#if __has_builtin(__builtin_amdgcn_exp2f)
    return __builtin_amdgcn_exp2f(x);   // v_exp_f32
#else
    return exp2f(x);
#endif
}

// ---- kernel 0: zero the 45 global accumulators ----------------------------
__global__ void ece_zero_kernel(float* __restrict__ gbins) {
    int t = threadIdx.x;
    if (t < 3 * NBINS) gbins[t] = 0.0f;
}

// ---- kernel 1: streaming histogram ----------------------------------------
__global__ __launch_bounds__(TPB) void ece_hist_kernel(
    const float* __restrict__ logits,
    const int*   __restrict__ labels,
    float*       __restrict__ gbins,
    int n, int ntiles)
{
    // double-buffered tile: 2 x (192 rows x 144 B padded) = 54 KB
    __shared__ __align__(16) char bufc[2][BUFBYTES];
    __shared__ float whist[NWAVES * HSTRIDE];

    const int t    = threadIdx.x;
    const int wave = t >> 5;

    for (int i = t; i < NWAVES * HSTRIDE; i += TPB) whist[i] = 0.0f;
    __syncthreads();

    // LDS byte address of buf (generic shared ptr low 32 bits == LDS offset)
    const unsigned  ldsrow0 = (unsigned)(size_t)(&bufc[0][0]) + (unsigned)t * ROWPAD;
    const long long nrows   = (long long)n;
    const float     L2E     = 1.44269504088896340736f;

    int tile = blockIdx.x;

    // prologue: async-fetch first tile
    {
        long long r = (long long)tile * TPB + t; if (r >= nrows) r = nrows - 1;
        issue_row_copy(ldsrow0, logits + r * CLS);
    }

    int bufidx = 0;
    for (; tile < ntiles; tile += gridDim.x, bufidx ^= 1) {
        const long long row   = (long long)tile * TPB + t;
        const bool      valid = row < nrows;
        const int       label = valid ? labels[row] : 0;

        // prefetch next tile into other buffer, then wait for current tile
        // (async loads complete in order: <=8 outstanding == next tile only)
        const int ntile = tile + gridDim.x;
        if (ntile < ntiles) {
            long long r = (long long)ntile * TPB + t; if (r >= nrows) r = nrows - 1;
            issue_row_copy(ldsrow0 + (unsigned)((bufidx ^ 1) * BUFBYTES), logits + r * CLS);
            wait_asynccnt<8>();
        } else {
            wait_asynccnt<0>();
        }

        // this thread's row from LDS: 8x ds_load_b128, immediate offsets
        const char*   rowp = &bufc[bufidx][0] + t * ROWPAD;
        const float4* p    = (const float4*)rowp;
        float x[CLS];
        #pragma unroll
        for (int j = 0; j < 8; ++j) {
            float4 q = p[j];
            x[4 * j + 0] = q.x; x[4 * j + 1] = q.y;
            x[4 * j + 2] = q.z; x[4 * j + 3] = q.w;
        }

        // tree max (depth 5, max3-fusable)
        float m[8];
        #pragma unroll
        for (int j = 0; j < 8; ++j)
            m[j] = fmaxf(fmaxf(x[4 * j], x[4 * j + 1]), fmaxf(x[4 * j + 2], x[4 * j + 3]));
        const float mA = fmaxf(fmaxf(m[0], m[1]), fmaxf(m[2], m[3]));
        const float mB = fmaxf(fmaxf(m[4], m[5]), fmaxf(m[6], m[7]));
        const float mx = fmaxf(mA, mB);

        // sum exp(x - mx) = exp2(fma(x, log2e, -mx*log2e)); 4-way ILP
        const float nm = -mx * L2E;
        float s0 = 0.f, s1 = 0.f, s2 = 0.f, s3 = 0.f;
        #pragma unroll
        for (int i = 0; i < CLS; i += 4) {
            s0 += fast_exp2(fmaf(x[i + 0], L2E, nm));
            s1 += fast_exp2(fmaf(x[i + 1], L2E, nm));
            s2 += fast_exp2(fmaf(x[i + 2], L2E, nm));
            s3 += fast_exp2(fmaf(x[i + 3], L2E, nm));
        }
        const float conf = 1.0f / ((s0 + s1) + (s2 + s3));   // max softmax prob

        // accuracy: logits[row][label] == row max  <=>  pred == label
        const int   lbl = label & (CLS - 1);
        const float xl  = *(const float*)(rowp + lbl * 4);

        if (valid) {
            const float accv = ((xl == mx) && (label == lbl)) ? 1.0f : 0.0f;
            int b = (int)ceilf(conf * (float)NBINS) - 1;
            b = b < 0 ? 0 : (b > NBINS - 1 ? NBINS - 1 : b);
            float* h = &whist[wave * HSTRIDE];
            atomicAdd(&h[b],      1.0f);    // ds_add_f32
            atomicAdd(&h[16 + b], conf);
            atomicAdd(&h[32 + b], accv);
        }
    }

    __syncthreads();
    if (t < 3 * NBINS) {
        const int grp = t / NBINS;
        const int b   = t - grp * NBINS;
        const int off = (grp == 0 ? 0 : (grp == 1 ? 16 : 32)) + b;
        float sum = 0.0f;
        #pragma unroll
        for (int w = 0; w < NWAVES; ++w) sum += whist[w * HSTRIDE + off];
        atomicAdd(&gbins[grp * NBINS + b], sum);   // global_atomic_add_f32
    }
}

// ---- kernel 2: final 15-bin ECE reduction ---------------------------------
__global__ void ece_final_kernel(const float* __restrict__ gbins,
                                 float* __restrict__ out, float n_total)
{
    if (threadIdx.x == 0) {
        float e = 0.0f;
        for (int b = 0; b < NBINS; ++b) {
            const float c  = gbins[b];
            const float cs = gbins[NBINS + b];
            const float as = gbins[2 * NBINS + b];
            if (c > 0.0f) {
                const float sc = fmaxf(c, 1.0f);
                e += fabsf(cs / sc - as / sc) * (c / n_total);
            }
        }
        out[0] = e;
    }
}

extern "C" void kernel_launch(void* const* d_in, const int* in_sizes, int n_in,
                              void* d_out, int out_size, void* d_ws, size_t ws_size,
                              hipStream_t stream) {
    const float* logits = (const float*)d_in[0];
    const int*   labels = (const int*)d_in[1];
    float*       gbins  = (float*)d_ws;      // 45 floats: count[15], conf[15], acc[15]
    float*       out    = (float*)d_out;

    const int n      = in_sizes[1];
    const int ntiles = (n + TPB - 1) / TPB;
    int blocks = ntiles < 1280 ? ntiles : 1280;

    ece_zero_kernel <<<1,      64,  0, stream>>>(gbins);
    ece_hist_kernel <<<blocks, TPB, 0, stream>>>(logits, labels, gbins, n, ntiles);
    ece_final_kernel<<<1,      32,  0, stream>>>(gbins, out, (float)n);
}